// SDFNetwork_34926674051658
// MI455X (gfx1250) — compile-verified
//
#include <hip/hip_runtime.h>

// ---------------------------------------------------------------------------
// GridEncoder static config (compile-time; matches reference exactly):
//   NUM_LEVELS=8, base_res=8, desired_res=32 -> growth g = 4^(1/7)
//   scale_l = 8*g^l - 1 ; side_l = ceil(scale_l)+2 ; offsets = cumsum(side^3)
//   All levels dense (side^3 < 2^18), total params = 85268.
// ---------------------------------------------------------------------------
__constant__ __device__ const float kScale[8] = {
    7.0f, 8.7521092f, 10.8879542f, 13.4915786f,
    16.6654321f, 20.5344028f, 25.2507310f, 31.0f};
__constant__ __device__ const int kSide[8] = {9, 11, 13, 16, 19, 23, 28, 33};
__constant__ __device__ const int kOff[8]  = {0, 729, 2060, 4257,
                                              8353, 15212, 27379, 49331};

typedef __attribute__((ext_vector_type(2))) float v2f;
typedef __attribute__((ext_vector_type(8))) float v8f;

#define HSTRIDE 18  // padded row stride (floats); 72B keeps v2f 8B-aligned

__global__ __launch_bounds__(256)
void ngp_encode_mlp_kernel(const float* __restrict__ x,
                           const float* __restrict__ table,
                           const float* __restrict__ W0,
                           const float* __restrict__ W1,
                           const float* __restrict__ W2,
                           float* __restrict__ out, int N) {
    // Per-wave staging buffer: 32 points x 16 feats (padded).
    __shared__ float Hs[8][32 * HSTRIDE];
    const int lane = threadIdx.x & 31;
    const int wave = threadIdx.x >> 5;
    float* Hw = Hs[wave];

    int n = blockIdx.x * 256 + threadIdx.x;
    if (n >= N) n = N - 1;  // clamp (no branch-out) so EXEC stays all-ones for WMMA

    const float px = x[3 * n + 0];
    const float py = x[3 * n + 1];
    const float pz = x[3 * n + 2];

    // ---------------- dense grid encode: 8 levels x 2 feats -----------------
    float h[16];
#pragma unroll
    for (int l = 0; l < 8; ++l) {
        const float sc = kScale[l];
        const int side = kSide[l];
        const int side2 = side * side;
        const float fx = px * sc + 0.5f;
        const float fy = py * sc + 0.5f;
        const float fz = pz * sc + 0.5f;
        const float gx = floorf(fx), gy = floorf(fy), gz = floorf(fz);
        const float tx = fx - gx, ty = fy - gy, tz = fz - gz;
        const int ix = (int)gx, iy = (int)gy, iz = (int)gz;
        const int base = kOff[l] + ix + iy * side + iz * side2;

        float f0 = 0.0f, f1 = 0.0f;
#pragma unroll
        for (int dz = 0; dz < 2; ++dz) {
            const float wz = dz ? tz : (1.0f - tz);
#pragma unroll
            for (int dy = 0; dy < 2; ++dy) {
                const float wyz = wz * (dy ? ty : (1.0f - ty));
                const int idx = base + dy * side + dz * side2;
                // corners (0,dy,dz) and (1,dy,dz) are adjacent table rows
                const float* t = table + 2 * idx;
                const v2f c0 = *(const v2f*)(t);       // row idx
                const v2f c1 = *(const v2f*)(t + 2);   // row idx+1
                const float w0 = wyz * (1.0f - tx);
                const float w1 = wyz * tx;
                f0 += w0 * c0.x + w1 * c1.x;
                f1 += w0 * c0.y + w1 * c1.y;
            }
        }
        h[2 * l + 0] = f0;
        h[2 * l + 1] = f1;
    }

    // Stage features to LDS: H[point][feat]
#pragma unroll
    for (int k = 0; k < 16; k += 2) {
        v2f tmp = {h[k], h[k + 1]};
        *(v2f*)(Hw + lane * HSTRIDE + k) = tmp;
    }
    __syncthreads();

    // -------------------- MLP: two 16x16 WMMA layers + ReLU ------------------
    // A frag (f32 16x4): lane(half=lane>>4, m=lane&15) holds H[m][4c+2*half +{0,1}]
    // B frag (f32 4x16): lane holds B[4c+2*half +{0,1}][m] = W[m][4c+2*half +{0,1}]
    // D (f32 16x16):     VGPR v, lane -> D[v + 8*half][m]
    const int half = lane >> 4;
    const int m = lane & 15;

#pragma unroll
    for (int layer = 0; layer < 2; ++layer) {
        const float* __restrict__ W = layer ? W1 : W0;
        v2f b[4];
#pragma unroll
        for (int c = 0; c < 4; ++c)
            b[c] = *(const v2f*)(W + m * 16 + 4 * c + 2 * half);

        v8f acc0 = {0.f, 0.f, 0.f, 0.f, 0.f, 0.f, 0.f, 0.f};
        v8f acc1 = acc0;
#pragma unroll
        for (int c = 0; c < 4; ++c) {
            const v2f a0 = *(const v2f*)(Hw + m * HSTRIDE + 4 * c + 2 * half);
            const v2f a1 = *(const v2f*)(Hw + (16 + m) * HSTRIDE + 4 * c + 2 * half);
            acc0 = __builtin_amdgcn_wmma_f32_16x16x4_f32(
                false, a0, false, b[c], (short)0, acc0, false, false);
            acc1 = __builtin_amdgcn_wmma_f32_16x16x4_f32(
                false, a1, false, b[c], (short)0, acc1, false, false);
        }
        __syncthreads();  // all A-frag reads done before overwriting H
#pragma unroll
        for (int v = 0; v < 8; ++v) {
            const float r0 = acc0[v] > 0.0f ? acc0[v] : 0.0f;  // ReLU
            const float r1 = acc1[v] > 0.0f ? acc1[v] : 0.0f;
            Hw[(v + 8 * half) * HSTRIDE + m] = r0;
            Hw[(16 + v + 8 * half) * HSTRIDE + m] = r1;
        }
        __syncthreads();
    }

    // ----------------------------- final 16 -> 1 -----------------------------
    float o = 0.0f;
#pragma unroll
    for (int k = 0; k < 16; ++k)
        o += Hw[lane * HSTRIDE + k] * W2[k];
    out[n] = o;
}

extern "C" void kernel_launch(void* const* d_in, const int* in_sizes, int n_in,
                              void* d_out, int out_size, void* d_ws, size_t ws_size,
                              hipStream_t stream) {
    const float* x     = (const float*)d_in[0];  // [N,3]
    const float* table = (const float*)d_in[1];  // [85268,2]
    const float* W0    = (const float*)d_in[2];  // [16,16]
    const float* W1    = (const float*)d_in[3];  // [16,16]
    const float* W2    = (const float*)d_in[4];  // [1,16]
    float* out = (float*)d_out;                  // [N,1]

    const int N = in_sizes[0] / 3;
    const int blocks = (N + 255) / 256;
    ngp_encode_mlp_kernel<<<blocks, 256, 0, stream>>>(x, table, W0, W1, W2, out, N);
}